// SAGENetwork_49950469652898
// MI455X (gfx1250) — compile-verified
//
#include <hip/hip_runtime.h>
#include <hip/hip_bf16.h>

// ---------------- problem constants (from reference) ----------------
#define NN      50000      // N_NODES
#define NE      500000     // N_EDGES
#define FIN     64
#define HID     128
#define NB      50         // graphs
#define NPER    1000       // nodes per graph
#define NEG     0.01f      // leaky slope
#define NHEAD   5
#define NCLUS   10

typedef float v2f __attribute__((ext_vector_type(2)));
typedef float v8f __attribute__((ext_vector_type(8)));

// ---------------- workspace layout (float offsets) ----------------
#define OFF_AGG1    ((size_t)0)                       // [NN*FIN]
#define OFF_CNT     (OFF_AGG1 + (size_t)NN*FIN)       // [NN]
#define OFF_H1      (OFF_CNT  + (size_t)NN)           // [NN*HID]
#define OFF_AGG2    (OFF_H1   + (size_t)NN*HID)       // [NN*HID]
#define OFF_H2      (OFF_AGG2 + (size_t)NN*HID)       // [NN*HID]
#define OFF_S1      (OFF_H2   + (size_t)NN*HID)       // [NN*NCLUS]
#define OFF_POOL1   (OFF_S1   + (size_t)NN*NCLUS)     // [NB*NCLUS*HID]
#define OFF_P1      (OFF_POOL1+ (size_t)NB*NCLUS*HID) // [NB*NCLUS*HID]
#define OFF_K2S     (OFF_P1   + (size_t)NB*NCLUS*HID) // [NHEAD*NCLUS]

__device__ __forceinline__ float leaky(float v) { return v > 0.f ? v : NEG * v; }

// ---------------- edge scatter: agg[dst] += feat[src], cnt[dst] += 1 ----
// one thread per (edge, float4): b128 gather + 4 f32 atomics
template<int F>
__global__ void scatter_add_kernel(const float* __restrict__ feat,
                                   const int* __restrict__ ei,
                                   float* __restrict__ agg,
                                   float* __restrict__ cnt,
                                   int count_too) {
  constexpr int G = F / 4;                  // float4 groups per row
  int tid = blockIdx.x * blockDim.x + threadIdx.x;
  if (tid >= NE * G) return;
  int e = tid / G;
  int g = tid - e * G;
  int f = g * 4;
  int src = ei[e];
  int dst = ei[NE + e];
  float4 v = *(const float4*)(feat + (size_t)src * F + f);
  float* d = agg + (size_t)dst * F + f;
  atomicAdd(d + 0, v.x);
  atomicAdd(d + 1, v.y);
  atomicAdd(d + 2, v.z);
  atomicAdd(d + 3, v.w);
  if (count_too && g == 0) atomicAdd(&cnt[dst], 1.0f);
}

// ---------------- dual fp32 WMMA GEMM: LDS-staged A, fused mean/bias/leaky ---
// out[m,n] = leaky( (A0[m,:]/max(cnt[m],1)) @ W0[n,:] + A1[m,:] @ W1[n,:] + bias[n] )
// grid.x = NN/16 row strips; block = 256 (8 waves -> 8 column tiles of 16); N = HID
template<int K>
__global__ void gemm_dual_wmma(const float* __restrict__ A0,
                               const float* __restrict__ A1,
                               const float* __restrict__ W0,
                               const float* __restrict__ W1,
                               const float* __restrict__ cnt,
                               const float* __restrict__ bias,
                               float* __restrict__ out) {
  constexpr int LSTR = K + 4;               // LDS row stride: = 4 mod 64 -> conflict-free
  __shared__ float lA[16 * LSTR];

  const int tid  = threadIdx.x;
  const int wave = tid >> 5;
  const int lane = tid & 31;
  const int half = lane >> 4;               // 0: K pair {0,1}, 1: K pair {2,3}
  const int mn   = lane & 15;               // m for A-frag, n for B/C-frag
  const int row0 = blockIdx.x * 16;
  const int n0   = wave * 16;

  v8f c = {};
  #pragma unroll
  for (int pass = 0; pass < 2; ++pass) {
    const float* __restrict__ A = pass ? A1 : A0;
    const float* __restrict__ W = pass ? W1 : W0;

    // cooperative stage: 16 rows x K floats via float4, scaled by 1/deg on pass 0
    constexpr int VEC = K / 4;              // float4s per row
    constexpr int TOT = 16 * VEC;           // 256 (K=64) or 512 (K=128)
    #pragma unroll
    for (int i = tid; i < TOT; i += 256) {
      int r  = i / VEC;
      int cc = (i - r * VEC) * 4;
      float4 v = *(const float4*)(A + (size_t)(row0 + r) * K + cc);
      float s = (pass == 0) ? (1.f / fmaxf(cnt[row0 + r], 1.f)) : 1.f;
      float* dp = &lA[r * LSTR + cc];
      dp[0] = v.x * s; dp[1] = v.y * s; dp[2] = v.z * s; dp[3] = v.w * s;
    }
    __syncthreads();

    const float* wrow = W + (size_t)(n0 + mn) * K;
    const float* arow = &lA[mn * LSTR];
    #pragma unroll
    for (int kb = 0; kb < K; kb += 4) {
      const int k = kb + half * 2;
      v2f a, b;
      a.x = arow[k];  a.y = arow[k + 1];     // ds_load_b64, conflict-free
      b.x = wrow[k];  b.y = wrow[k + 1];     // B[k][n] = W[n][k]  (out = in @ W^T)
      c = __builtin_amdgcn_wmma_f32_16x16x4_f32(false, a, false, b,
                                                (short)0, c, false, false);
    }
    __syncthreads();
  }

  const float bj = bias[n0 + mn];
  #pragma unroll
  for (int v = 0; v < 8; ++v) {
    const int m = v + half * 8;
    out[(size_t)(row0 + m) * HID + n0 + mn] = leaky(c[v] + bj);
  }
}

// ---------------- k1 squared row norms (50 rows of 128) ----------------
__global__ void k2sum_kernel(const float* __restrict__ k1, float* __restrict__ k2s) {
  int idx = blockIdx.x * blockDim.x + threadIdx.x;
  if (idx >= NHEAD * NCLUS) return;
  float s = 0.f;
  for (int f = 0; f < HID; ++f) { float v = k1[idx * HID + f]; s += v * v; }
  k2s[idx] = s;
}

// ---------------- MemPool-1 soft assignment S [NN, NCLUS] ----------------
// student-t (tau=1): dist = 1/(1+d); normalize over clusters per head;
// S = softmax_k( sum_h conv_w[h] * dist_norm[h,k] )
__global__ void assign_kernel(const float* __restrict__ X,      // h2 [NN, HID]
                              const float* __restrict__ k1,     // [50, HID]
                              const float* __restrict__ k2s,    // [50]
                              const float* __restrict__ convw,  // [5]
                              float* __restrict__ S1) {         // [NN, NCLUS]
  int node = blockIdx.x * blockDim.x + threadIdx.x;
  if (node >= NN) return;
  const float* xr = X + (size_t)node * HID;

  float x2 = 0.f;
  float dot[NHEAD][NCLUS];
  #pragma unroll
  for (int h = 0; h < NHEAD; ++h)
    #pragma unroll
    for (int k = 0; k < NCLUS; ++k) dot[h][k] = 0.f;

  for (int f = 0; f < HID; ++f) {
    float xv = xr[f];
    x2 += xv * xv;
    #pragma unroll
    for (int h = 0; h < NHEAD; ++h)
      #pragma unroll
      for (int k = 0; k < NCLUS; ++k)
        dot[h][k] += xv * k1[(h * NCLUS + k) * HID + f];   // uniform -> scalar loads
  }

  float S[NCLUS];
  #pragma unroll
  for (int k = 0; k < NCLUS; ++k) S[k] = 0.f;

  #pragma unroll
  for (int h = 0; h < NHEAD; ++h) {
    float dh[NCLUS], dsum = 0.f;
    #pragma unroll
    for (int k = 0; k < NCLUS; ++k) {
      float d = fmaxf(x2 + k2s[h * NCLUS + k] - 2.f * dot[h][k], 0.f);
      dh[k] = 1.f / (1.f + d);              // (1 + d/tau)^(-(tau+1)/2), tau=1
      dsum += dh[k];
    }
    float cw = convw[h] / dsum;
    #pragma unroll
    for (int k = 0; k < NCLUS; ++k) S[k] += cw * dh[k];
  }

  float mx = S[0];
  #pragma unroll
  for (int k = 1; k < NCLUS; ++k) mx = fmaxf(mx, S[k]);
  float es = 0.f;
  #pragma unroll
  for (int k = 0; k < NCLUS; ++k) { S[k] = __expf(S[k] - mx); es += S[k]; }
  float inv = 1.f / es;
  #pragma unroll
  for (int k = 0; k < NCLUS; ++k) S1[(size_t)node * NCLUS + k] = S[k] * inv;
}

// ---------------- pooled[b,k,f] = sum_n S[b,n,k] * x[b,n,f] ----------------
// grid = (NB, NCLUS), block = HID threads (one f each); direct writes.
__global__ void pool1_kernel(const float* __restrict__ S1,
                             const float* __restrict__ X,    // h2 [NN, HID]
                             float* __restrict__ pooled) {   // [NB, NCLUS, HID]
  int b = blockIdx.x, kk = blockIdx.y, f = threadIdx.x;
  float acc = 0.f;
  size_t base = (size_t)b * NPER;
  for (int n = 0; n < NPER; ++n) {
    float s = S1[(base + n) * NCLUS + kk];   // uniform across block -> broadcast
    acc += s * X[(base + n) * HID + f];
  }
  pooled[((size_t)b * NCLUS + kk) * HID + f] = acc;
}

// ---------------- p1 = leaky(pooled @ lin1_w^T) ----------------
__global__ void lin1_kernel(const float* __restrict__ pooled,
                            const float* __restrict__ lin1w,  // [HID, HID]
                            float* __restrict__ p1) {
  int row = blockIdx.x, j = threadIdx.x;
  const float* pr = pooled + (size_t)row * HID;
  const float* wr = lin1w + (size_t)j * HID;
  float acc = 0.f;
  for (int f = 0; f < HID; ++f) acc += pr[f] * wr[f];
  p1[(size_t)row * HID + j] = leaky(acc);
}

// ---------------- MemPool-2 (K=1 -> S==1) + lin2: out[b,a] ----------------
// pooled2[b,f] = sum_{n<10} p1[b,n,f]; out[b,a] = pooled2[b,:] . lin2_w[a,:]
__global__ void final_kernel(const float* __restrict__ p1,
                             const float* __restrict__ lin2w,  // [8, HID]
                             float* __restrict__ out) {
  int tid = blockIdx.x * blockDim.x + threadIdx.x;
  if (tid >= NB * 8) return;
  int b = tid >> 3, a = tid & 7;
  const float* wr = lin2w + (size_t)a * HID;
  float acc = 0.f;
  for (int f = 0; f < HID; ++f) {
    float s = 0.f;
    #pragma unroll
    for (int n = 0; n < NCLUS; ++n) s += p1[((size_t)b * NCLUS + n) * HID + f];
    acc += s * wr[f];
  }
  out[tid] = acc;
}

// =====================================================================
extern "C" void kernel_launch(void* const* d_in, const int* in_sizes, int n_in,
                              void* d_out, int out_size, void* d_ws, size_t ws_size,
                              hipStream_t stream) {
  const float* x      = (const float*)d_in[0];
  const int*   ei     = (const int*)  d_in[1];
  // d_in[2] = batch (arange//NPER) -> dense reshape, unused
  const float* W1_l   = (const float*)d_in[3];
  const float* b1     = (const float*)d_in[4];
  const float* W1_r   = (const float*)d_in[5];
  const float* W2_l   = (const float*)d_in[6];
  const float* b2     = (const float*)d_in[7];
  const float* W2_r   = (const float*)d_in[8];
  const float* k1     = (const float*)d_in[9];
  const float* conv1w = (const float*)d_in[10];
  const float* lin1w  = (const float*)d_in[11];
  // d_in[12] = k2, d_in[13] = conv2_w: drop out exactly (softmax over 1 cluster == 1)
  const float* lin2w  = (const float*)d_in[14];
  float* out = (float*)d_out;

  float* ws     = (float*)d_ws;
  float* agg1   = ws + OFF_AGG1;
  float* cnt    = ws + OFF_CNT;
  float* h1     = ws + OFF_H1;
  float* agg2   = ws + OFF_AGG2;
  float* h2     = ws + OFF_H2;
  float* S1     = ws + OFF_S1;
  float* pooled = ws + OFF_POOL1;
  float* p1     = ws + OFF_P1;
  float* k2s    = ws + OFF_K2S;

  // zero accumulators (graph-capturable async memsets)
  hipMemsetAsync(agg1, 0, (size_t)NN * FIN * sizeof(float), stream);
  hipMemsetAsync(cnt,  0, (size_t)NN * sizeof(float), stream);
  hipMemsetAsync(agg2, 0, (size_t)NN * HID * sizeof(float), stream);

  // ---- SAGE layer 1 (mean fused into GEMM staging) ----
  {
    int total = NE * (FIN / 4);
    scatter_add_kernel<FIN><<<(total + 255) / 256, 256, 0, stream>>>(x, ei, agg1, cnt, 1);
    gemm_dual_wmma<FIN><<<NN / 16, 256, 0, stream>>>(agg1, x, W1_l, W1_r, cnt, b1, h1);
  }
  // ---- SAGE layer 2 ----
  {
    int total = NE * (HID / 4);
    scatter_add_kernel<HID><<<(total + 255) / 256, 256, 0, stream>>>(h1, ei, agg2, cnt, 0);
    gemm_dual_wmma<HID><<<NN / 16, 256, 0, stream>>>(agg2, h1, W2_l, W2_r, cnt, b2, h2);
  }
  // ---- MemPooling 1 ----
  k2sum_kernel<<<1, 64, 0, stream>>>(k1, k2s);
  assign_kernel<<<(NN + 127) / 128, 128, 0, stream>>>(h2, k1, k2s, conv1w, S1);
  pool1_kernel<<<dim3(NB, NCLUS), HID, 0, stream>>>(S1, h2, pooled);
  lin1_kernel<<<NB * NCLUS, HID, 0, stream>>>(pooled, lin1w, p1);
  // ---- MemPooling 2 (degenerate) + lin2 ----
  final_kernel<<<1, 512, 0, stream>>>(p1, lin2w, out);
}